// BlockNet_89713276879602
// MI455X (gfx1250) — compile-verified
//
#include <hip/hip_runtime.h>
#include <hip/hip_bf16.h>

// ---------------------------------------------------------------------------
// BlockNet for MI455X (gfx1250): fp32 WMMA (V_WMMA_F32_16X16X4_F32) GEMMs +
// flash-attention, wave32. Problem is memory bound (~40 GFLOP vs ~33MB I/O at
// 23.3 TB/s), so fp32 matrix path keeps reference numerics at no cost.
// ---------------------------------------------------------------------------

typedef float v2f __attribute__((ext_vector_type(2)));
typedef float v8f __attribute__((ext_vector_type(8)));

#define T_DIM 2
#define B_DIM 8
#define C_DIM 256
#define S_DIM 1024
#define SK_DIM 512
#define F_DIM 1024
#define H_DIM 8
#define HD_DIM 32
#define Q_LEN 2049   // T*S + 1 (cls)
#define K_LEN 1024   // T*SK

__device__ __forceinline__ v8f wmma_f32(v2f a, v2f b, v8f c) {
  return __builtin_amdgcn_wmma_f32_16x16x4_f32(
      /*neg_a=*/false, a, /*neg_b=*/false, b,
      /*c_mod=*/(short)0, c, /*reuse_a=*/false, /*reuse_b=*/false);
}

// ---------------------------------------------------------------------------
// Channel LayerNorm: x (slab, C=256, S=1024), gamma/beta indexed by t = slab/B
// ---------------------------------------------------------------------------
__global__ __launch_bounds__(256) void ln_c_kernel(
    const float* __restrict__ x, const float* __restrict__ g,
    const float* __restrict__ be, float* __restrict__ y) {
  const int slab = blockIdx.y;              // t*B + b
  const int t = slab / B_DIM;
  const int s = blockIdx.x * 256 + threadIdx.x;
  const float* xs = x + (size_t)slab * C_DIM * S_DIM + s;
  float sum = 0.f, sum2 = 0.f;
  for (int c = 0; c < C_DIM; ++c) {
    float v = xs[(size_t)c * S_DIM];
    sum += v; sum2 += v * v;
  }
  const float mean = sum * (1.f / C_DIM);
  const float var = sum2 * (1.f / C_DIM) - mean * mean;
  const float rstd = rsqrtf(var + 1e-5f);
  float* ys = y + (size_t)slab * C_DIM * S_DIM + s;
  for (int c = 0; c < C_DIM; ++c) {
    float v = (xs[(size_t)c * S_DIM] - mean) * rstd;
    ys[(size_t)c * S_DIM] = v * g[t * C_DIM + c] + be[t * C_DIM + c];
  }
}

// ---------------------------------------------------------------------------
// Row LayerNorm over last dim (256) for small (rows, 256) tensors (cls path)
// ---------------------------------------------------------------------------
__global__ __launch_bounds__(64) void rowln_kernel(
    const float* __restrict__ x, const float* __restrict__ g,
    const float* __restrict__ be, float* __restrict__ y, int rows) {
  int r = blockIdx.x * 64 + threadIdx.x;
  if (r >= rows) return;
  const float* xr = x + (size_t)r * C_DIM;
  float s = 0.f, s2 = 0.f;
  for (int c = 0; c < C_DIM; ++c) { float v = xr[c]; s += v; s2 += v * v; }
  float mean = s * (1.f / C_DIM);
  float var = s2 * (1.f / C_DIM) - mean * mean;
  float rstd = rsqrtf(var + 1e-5f);
  float* yr = y + (size_t)r * C_DIM;
  for (int c = 0; c < C_DIM; ++c)
    yr[c] = (xr[c] - mean) * rstd * g[c] + be[c];
}

// ---------------------------------------------------------------------------
// Depthwise conv (K=5, pad=2) + bias + BN-affine + ReLU.  Batched over t*B+b.
// ---------------------------------------------------------------------------
__global__ __launch_bounds__(256) void dw_kernel(
    const float* __restrict__ x, const float* __restrict__ w,
    const float* __restrict__ db, const float* __restrict__ bs,
    const float* __restrict__ bb, float* __restrict__ y,
    int Cn, int Sin, int So, int stride) {
  const int z = blockIdx.y;                 // t*B + b
  const int t = z / B_DIM;
  int i = blockIdx.x * 256 + threadIdx.x;
  if (i >= Cn * So) return;
  const int so = i % So;
  const int c = i / So;
  const float* xs = x + ((size_t)z * Cn + c) * Sin;
  const float* wc = w + ((size_t)t * Cn + c) * 5;
  float acc = 0.f;
  const int sbase = so * stride - 2;
#pragma unroll
  for (int k = 0; k < 5; ++k) {
    int si = sbase + k;
    if (si >= 0 && si < Sin) acc += wc[k] * xs[si];
  }
  acc += db[t * Cn + c];
  acc = acc * bs[t * Cn + c] + bb[t * Cn + c];
  y[((size_t)z * Cn + c) * So + so] = fmaxf(acc, 0.f);
}

// ---------------------------------------------------------------------------
// WMMA fp32 GEMM:  Y[z] = W[z/wdiv] (M,Kc) * X[z] (Kc,N) + bias (+R) [, relu]
// One wave computes a 16x32 output tile (two N-sub-tiles sharing the A
// fragment -> 2 WMMAs per A load); 8 waves/block tile 256 columns.
// ---------------------------------------------------------------------------
__global__ __launch_bounds__(256) void gemm_pw_kernel(
    const float* __restrict__ W, long wstride, int wdiv,
    const float* __restrict__ bias, long bstride,
    const float* __restrict__ X, long xstride,
    const float* __restrict__ R, long rstride,
    float* __restrict__ Y, long ystride,
    int M, int N, int Kc, int relu) {
  const int z = blockIdx.z;
  const int tw = z / wdiv;
  const float* Wz = W + (size_t)tw * wstride;
  const float* bz = bias + (size_t)tw * bstride;
  const float* Xz = X + (size_t)z * xstride;
  float* Yz = Y + (size_t)z * ystride;
  const float* Rz = R ? (R + (size_t)z * rstride) : (const float*)0;

  const int lane = threadIdx.x & 31;
  const int wv = threadIdx.x >> 5;
  const int half = lane >> 4;               // 0: K pair {0,1}, 1: {2,3}
  const int l16 = lane & 15;
  const int m0 = blockIdx.x * 16;
  const int n0 = blockIdx.y * 256 + wv * 32;

  const int nc0 = n0 + l16;
  const int nc1 = n0 + 16 + l16;
  const int ncl0 = (nc0 < N) ? nc0 : (N - 1);
  const int ncl1 = (nc1 < N) ? nc1 : (N - 1);

  const int koff = 2 * half;
  const float* Arow = Wz + (size_t)(m0 + l16) * Kc + koff;  // A: 16x4 layout
  const size_t stepB = (size_t)4 * N;
  const float* b0lo = Xz + (size_t)koff * N + ncl0;         // B: 4x16 layout
  const float* b0hi = b0lo + N;
  const float* b1lo = Xz + (size_t)koff * N + ncl1;
  const float* b1hi = b1lo + N;

  v8f acc0 = {0.f, 0.f, 0.f, 0.f, 0.f, 0.f, 0.f, 0.f};
  v8f acc1 = {0.f, 0.f, 0.f, 0.f, 0.f, 0.f, 0.f, 0.f};
  for (int k = 0; k < Kc; k += 4) {
    v2f a = *(const v2f*)Arow;
    Arow += 4;
    v2f b0; b0.x = *b0lo; b0.y = *b0hi;
    b0lo += stepB; b0hi += stepB;
    v2f b1; b1.x = *b1lo; b1.y = *b1hi;
    b1lo += stepB; b1hi += stepB;
    acc0 = wmma_f32(a, b0, acc0);
    acc1 = wmma_f32(a, b1, acc1);
  }

  const int mbase = m0 + half * 8;          // D: vgpr r -> row mbase+r
  if (nc0 < N) {
    for (int r = 0; r < 8; ++r) {
      const int m = mbase + r;
      float v = acc0[r] + bz[m];
      if (Rz) v += Rz[(size_t)m * N + nc0];
      if (relu) v = fmaxf(v, 0.f);
      Yz[(size_t)m * N + nc0] = v;
    }
  }
  if (nc1 < N) {
    for (int r = 0; r < 8; ++r) {
      const int m = mbase + r;
      float v = acc1[r] + bz[m];
      if (Rz) v += Rz[(size_t)m * N + nc1];
      if (relu) v = fmaxf(v, 0.f);
      Yz[(size_t)m * N + nc1] = v;
    }
  }
}

// ---------------------------------------------------------------------------
// Flash attention: one wave per (b, h, 16-query tile).
// q: (T,B,C,S) from q-sepconv, qcls: (B,C), kv: (T,B,2C,SK) [k=ch 0..255,
// v=ch 256..511].  Output avt: (B,C,Q) with c = h*hd + d.
// ---------------------------------------------------------------------------
__device__ __forceinline__ float ld_query(const float* __restrict__ q,
                                          const float* __restrict__ qcls,
                                          int b, int c, int qpos) {
  if (qpos == Q_LEN - 1) return qcls[b * C_DIM + c];
  const int t = qpos >> 10;
  const int s = qpos & (S_DIM - 1);
  return q[((((size_t)t * B_DIM + b) * C_DIM) + c) * S_DIM + s];
}

__global__ __launch_bounds__(32) void attn_kernel(
    const float* __restrict__ q, const float* __restrict__ qcls,
    const float* __restrict__ kv, float* __restrict__ avt) {
  __shared__ float P[16 * 17];
  const int lane = threadIdx.x;
  const int half = lane >> 4;
  const int l16 = lane & 15;
  const int qt = blockIdx.x;                // 0..128
  const int b = blockIdx.y >> 3;
  const int h = blockIdx.y & 7;
  const float scale = 0.17677669529663687f; // 1/sqrt(32)

  // Preload Q fragments (A layout, scaled): frag j covers d = 4j..4j+3
  int qpos = qt * 16 + l16;
  if (qpos > Q_LEN - 1) qpos = Q_LEN - 1;
  v2f afr[8];
#pragma unroll
  for (int j = 0; j < 8; ++j) {
    const int c0 = h * HD_DIM + 4 * j + 2 * half;
    afr[j].x = ld_query(q, qcls, b, c0, qpos) * scale;
    afr[j].y = ld_query(q, qcls, b, c0 + 1, qpos) * scale;
  }

  float mrow[8], lrow[8];
  v8f acc0 = {0.f, 0.f, 0.f, 0.f, 0.f, 0.f, 0.f, 0.f};
  v8f acc1 = {0.f, 0.f, 0.f, 0.f, 0.f, 0.f, 0.f, 0.f};
#pragma unroll
  for (int r = 0; r < 8; ++r) { mrow[r] = -1e30f; lrow[r] = 0.f; }

  for (int kt = 0; kt < K_LEN / 16; ++kt) {
    // ---- scores S = (q*scale)^T k : 8 WMMAs over hd=32 ----
    v8f st = {0.f, 0.f, 0.f, 0.f, 0.f, 0.f, 0.f, 0.f};
    const int kp = kt * 16 + l16;
    const int tk = kp >> 9;                 // branch index (tile never straddles)
    const int s2 = kp & (SK_DIM - 1);
    const float* kbase =
        kv + (((size_t)tk * B_DIM + b) * (2 * C_DIM) + h * HD_DIM) * SK_DIM + s2;
#pragma unroll
    for (int j = 0; j < 8; ++j) {
      const float* kr = kbase + (size_t)(4 * j + 2 * half) * SK_DIM;
      v2f bf;
      bf.x = kr[0];
      bf.y = kr[SK_DIM];                    // next channel (d+1), same key
      st = wmma_f32(afr[j], bf, st);
    }

    // ---- online softmax in C/D lane layout ----
#pragma unroll
    for (int r = 0; r < 8; ++r) {
      float v = st[r];
      v = fmaxf(v, __shfl_xor(v, 1, 16));
      v = fmaxf(v, __shfl_xor(v, 2, 16));
      v = fmaxf(v, __shfl_xor(v, 4, 16));
      v = fmaxf(v, __shfl_xor(v, 8, 16));
      const float mnew = fmaxf(mrow[r], v);
      const float corr = __expf(mrow[r] - mnew);
      const float p = __expf(st[r] - mnew);
      float ps = p;
      ps += __shfl_xor(ps, 1, 16);
      ps += __shfl_xor(ps, 2, 16);
      ps += __shfl_xor(ps, 4, 16);
      ps += __shfl_xor(ps, 8, 16);
      lrow[r] = lrow[r] * corr + ps;
      mrow[r] = mnew;
      acc0[r] *= corr;
      acc1[r] *= corr;
      st[r] = p;
    }

    // ---- restage P through LDS into A layout ----
#pragma unroll
    for (int r = 0; r < 8; ++r)
      P[(r + half * 8) * 17 + l16] = st[r];
    __syncthreads();

    // ---- acc += P * V^T : two 16x16 N-tiles (d 0..15, 16..31) ----
    const size_t vbase =
        (((size_t)tk * B_DIM + b) * (2 * C_DIM) + C_DIM + h * HD_DIM) * SK_DIM;
#pragma unroll
    for (int kk = 0; kk < 16; kk += 4) {
      v2f pa;
      pa.x = P[l16 * 17 + kk + 2 * half];
      pa.y = P[l16 * 17 + kk + 2 * half + 1];
      const int kp0 = kt * 16 + kk + 2 * half;
      const int s20 = kp0 & (SK_DIM - 1);
      const float* vr0 = kv + vbase + (size_t)l16 * SK_DIM + s20;  // d = l16
      v2f vb0; vb0.x = vr0[0]; vb0.y = vr0[1];
      const float* vr1 = vr0 + (size_t)16 * SK_DIM;                // d = 16+l16
      v2f vb1; vb1.x = vr1[0]; vb1.y = vr1[1];
      acc0 = wmma_f32(pa, vb0, acc0);
      acc1 = wmma_f32(pa, vb1, acc1);
    }
    __syncthreads();
  }

  // ---- normalize and store avt (B,C,Q) ----
#pragma unroll
  for (int r = 0; r < 8; ++r) {
    const int qrow = qt * 16 + half * 8 + r;
    if (qrow < Q_LEN) {
      const float inv = 1.f / lrow[r];
      const int c0 = h * HD_DIM + l16;
      avt[((size_t)b * C_DIM + c0) * Q_LEN + qrow] = acc0[r] * inv;
      avt[((size_t)b * C_DIM + 16 + c0) * Q_LEN + qrow] = acc1[r] * inv;
    }
  }
}

// ---------------------------------------------------------------------------
// x2 = xn + attn_tf  (in place into xn).  aout is (B,C,Q).
// ---------------------------------------------------------------------------
__global__ __launch_bounds__(256) void addattn_kernel(
    float* __restrict__ xn, const float* __restrict__ aout) {
  size_t i = (size_t)blockIdx.x * 256 + threadIdx.x;
  const int s = (int)(i & (S_DIM - 1));
  const int c = (int)((i >> 10) & (C_DIM - 1));
  const int b = (int)((i >> 18) & (B_DIM - 1));
  const int t = (int)(i >> 21);
  xn[i] += aout[(((size_t)b * C_DIM + c) * Q_LEN) + t * S_DIM + s];
}

// ---------------------------------------------------------------------------
// CLS helpers
// ---------------------------------------------------------------------------
__global__ __launch_bounds__(256) void xc_kernel(
    const float* __restrict__ clsn, const float* __restrict__ aout,
    float* __restrict__ xc) {
  int i = blockIdx.x * 256 + threadIdx.x;   // b*256 + c
  if (i >= B_DIM * C_DIM) return;
  xc[i] = clsn[i] + aout[(size_t)i * Q_LEN + (Q_LEN - 1)];
}

// Y[b,o] = relu?(bias[o] + sum_k X[b,k]*W[o,k]) (+res[b,o])
__global__ __launch_bounds__(256) void matvec_kernel(
    const float* __restrict__ X, const float* __restrict__ W,
    const float* __restrict__ bias, const float* __restrict__ res,
    float* __restrict__ Y, int Bn, int O, int Kc, int relu) {
  int i = blockIdx.x * 256 + threadIdx.x;
  if (i >= Bn * O) return;
  const int o = i % O;
  const int b = i / O;
  const float* xr = X + (size_t)b * Kc;
  const float* wr = W + (size_t)o * Kc;
  float acc = bias[o];
  for (int k = 0; k < Kc; ++k) acc += xr[k] * wr[k];
  if (relu) acc = fmaxf(acc, 0.f);
  if (res) acc += res[i];
  Y[i] = acc;
}

// ---------------------------------------------------------------------------
// Launch
// ---------------------------------------------------------------------------
extern "C" void kernel_launch(void* const* d_in, const int* in_sizes, int n_in,
                              void* d_out, int out_size, void* d_ws,
                              size_t ws_size, hipStream_t stream) {
  (void)in_sizes; (void)n_in; (void)out_size; (void)ws_size;
  const float* inp      = (const float*)d_in[0];
  const float* cls      = (const float*)d_in[1];
  const float* ln1_g    = (const float*)d_in[2];
  const float* ln1_b    = (const float*)d_in[3];
  const float* q_dw_w   = (const float*)d_in[4];
  const float* q_dw_b   = (const float*)d_in[5];
  const float* q_bn_s   = (const float*)d_in[6];
  const float* q_bn_b   = (const float*)d_in[7];
  const float* q_pw_w   = (const float*)d_in[8];
  const float* q_pw_b   = (const float*)d_in[9];
  const float* kv_dw_w  = (const float*)d_in[10];
  const float* kv_dw_b  = (const float*)d_in[11];
  const float* kv_bn_s  = (const float*)d_in[12];
  const float* kv_bn_b  = (const float*)d_in[13];
  const float* kv_pw_w  = (const float*)d_in[14];
  const float* kv_pw_b  = (const float*)d_in[15];
  const float* wq_cls   = (const float*)d_in[16];
  const float* bq_cls   = (const float*)d_in[17];
  const float* ln1c_g   = (const float*)d_in[18];
  const float* ln1c_b   = (const float*)d_in[19];
  const float* wout     = (const float*)d_in[20];
  const float* bout     = (const float*)d_in[21];
  const float* ff1_dw_w = (const float*)d_in[22];
  const float* ff1_dw_b = (const float*)d_in[23];
  const float* ff1_bn_s = (const float*)d_in[24];
  const float* ff1_bn_b = (const float*)d_in[25];
  const float* ff1_pw_w = (const float*)d_in[26];
  const float* ff1_pw_b = (const float*)d_in[27];
  const float* ff2_dw_w = (const float*)d_in[28];
  const float* ff2_dw_b = (const float*)d_in[29];
  const float* ff2_bn_s = (const float*)d_in[30];
  const float* ff2_bn_b = (const float*)d_in[31];
  const float* ff2_pw_w = (const float*)d_in[32];
  const float* ff2_pw_b = (const float*)d_in[33];
  const float* ln2_g    = (const float*)d_in[34];
  const float* ln2_b    = (const float*)d_in[35];
  const float* ln2c_g   = (const float*)d_in[36];
  const float* ln2c_b   = (const float*)d_in[37];
  const float* wff1     = (const float*)d_in[38];
  const float* bff1     = (const float*)d_in[39];
  const float* wff2     = (const float*)d_in[40];
  const float* bff2     = (const float*)d_in[41];

  float* out = (float*)d_out;
  float* ws = (float*)d_ws;

  const size_t TBCS = (size_t)T_DIM * B_DIM * C_DIM * S_DIM;   // 4,194,304
  const size_t TBFS = (size_t)T_DIM * B_DIM * F_DIM * S_DIM;   // 16,777,216
  const size_t BCQ  = (size_t)B_DIM * C_DIM * Q_LEN;           // 4,196,352

  size_t o = 0;
  float* xn    = ws + o; o += TBCS;        // xn, later x2 (in place)
  float* dwbuf = ws + o; o += TBFS;        // depthwise scratch (max F*S slabs)
  float* qbuf  = ws + o; o += TBCS;        // q, later ln2(x2)
  float* kvbuf = ws + o; o += (size_t)T_DIM * B_DIM * 2 * C_DIM * SK_DIM;
  float* avt   = ws + o; o += BCQ;
  float* aout  = ws + o; o += BCQ;
  float* h1    = ws + o; o += TBFS;
  float* clsn  = ws + o; o += B_DIM * C_DIM;
  float* qclsb = ws + o; o += B_DIM * C_DIM;
  float* xcb   = ws + o; o += B_DIM * C_DIM;
  float* lncx  = ws + o; o += B_DIM * C_DIM;
  float* h1c   = ws + o; o += B_DIM * F_DIM;

  const int TB = T_DIM * B_DIM;

  // 1) xn = ln_c(inp)
  ln_c_kernel<<<dim3(S_DIM / 256, TB), 256, 0, stream>>>(inp, ln1_g, ln1_b, xn);
  // 2) cls_norm
  rowln_kernel<<<1, 64, 0, stream>>>(cls, ln1c_g, ln1c_b, clsn, B_DIM);
  // 3) q_cls = cls_norm @ wq_cls.T + bq_cls
  matvec_kernel<<<(B_DIM * C_DIM + 255) / 256, 256, 0, stream>>>(
      clsn, wq_cls, bq_cls, (const float*)0, qclsb, B_DIM, C_DIM, C_DIM, 0);
  // 4) q depthwise
  dw_kernel<<<dim3((C_DIM * S_DIM) / 256, TB), 256, 0, stream>>>(
      xn, q_dw_w, q_dw_b, q_bn_s, q_bn_b, dwbuf, C_DIM, S_DIM, S_DIM, 1);
  // 5) q pointwise (WMMA): M=256, N=1024, K=256
  gemm_pw_kernel<<<dim3(C_DIM / 16, S_DIM / 256, TB), 256, 0, stream>>>(
      q_pw_w, (long)C_DIM * C_DIM, B_DIM, q_pw_b, (long)C_DIM,
      dwbuf, (long)C_DIM * S_DIM, (const float*)0, 0L,
      qbuf, (long)C_DIM * S_DIM, C_DIM, S_DIM, C_DIM, 0);
  // 6) kv depthwise (stride 2)
  dw_kernel<<<dim3((C_DIM * SK_DIM) / 256, TB), 256, 0, stream>>>(
      xn, kv_dw_w, kv_dw_b, kv_bn_s, kv_bn_b, dwbuf, C_DIM, S_DIM, SK_DIM, 2);
  // 7) kv pointwise: M=512, N=512, K=256
  gemm_pw_kernel<<<dim3((2 * C_DIM) / 16, SK_DIM / 256, TB), 256, 0, stream>>>(
      kv_pw_w, (long)2 * C_DIM * C_DIM, B_DIM, kv_pw_b, (long)2 * C_DIM,
      dwbuf, (long)C_DIM * SK_DIM, (const float*)0, 0L,
      kvbuf, (long)2 * C_DIM * SK_DIM, 2 * C_DIM, SK_DIM, C_DIM, 0);
  // 8) flash attention -> avt (B,C,Q)
  attn_kernel<<<dim3((Q_LEN + 15) / 16, B_DIM * H_DIM), 32, 0, stream>>>(
      qbuf, qclsb, kvbuf, avt);
  // 9) attn_out = wout @ avt + bout   (shared weight, batch over B)
  gemm_pw_kernel<<<dim3(C_DIM / 16, (Q_LEN + 255) / 256, B_DIM), 256, 0, stream>>>(
      wout, 0L, 1, bout, 0L,
      avt, (long)C_DIM * Q_LEN, (const float*)0, 0L,
      aout, (long)C_DIM * Q_LEN, C_DIM, Q_LEN, C_DIM, 0);
  // 10) x2 = xn + attn_tf (in place)
  addattn_kernel<<<(unsigned)(TBCS / 256), 256, 0, stream>>>(xn, aout);
  // 11) CLS path
  xc_kernel<<<(B_DIM * C_DIM + 255) / 256, 256, 0, stream>>>(clsn, aout, xcb);
  rowln_kernel<<<1, 64, 0, stream>>>(xcb, ln2c_g, ln2c_b, lncx, B_DIM);
  matvec_kernel<<<(B_DIM * F_DIM + 255) / 256, 256, 0, stream>>>(
      lncx, wff1, bff1, (const float*)0, h1c, B_DIM, F_DIM, C_DIM, 1);
  matvec_kernel<<<(B_DIM * C_DIM + 255) / 256, 256, 0, stream>>>(
      h1c, wff2, bff2, xcb, out + TBCS, B_DIM, C_DIM, F_DIM, 0);
  // 12) ln2_c(x2) -> qbuf
  ln_c_kernel<<<dim3(S_DIM / 256, TB), 256, 0, stream>>>(xn, ln2_g, ln2_b, qbuf);
  // 13) ff1 depthwise
  dw_kernel<<<dim3((C_DIM * S_DIM) / 256, TB), 256, 0, stream>>>(
      qbuf, ff1_dw_w, ff1_dw_b, ff1_bn_s, ff1_bn_b, dwbuf, C_DIM, S_DIM, S_DIM, 1);
  // 14) ff1 pointwise + outer relu: M=1024, N=1024, K=256
  gemm_pw_kernel<<<dim3(F_DIM / 16, S_DIM / 256, TB), 256, 0, stream>>>(
      ff1_pw_w, (long)F_DIM * C_DIM, B_DIM, ff1_pw_b, (long)F_DIM,
      dwbuf, (long)C_DIM * S_DIM, (const float*)0, 0L,
      h1, (long)F_DIM * S_DIM, F_DIM, S_DIM, C_DIM, 1);
  // 15) ff2 depthwise (F channels)
  dw_kernel<<<dim3((F_DIM * S_DIM) / 256, TB), 256, 0, stream>>>(
      h1, ff2_dw_w, ff2_dw_b, ff2_bn_s, ff2_bn_b, dwbuf, F_DIM, S_DIM, S_DIM, 1);
  // 16) ff2 pointwise + residual x2 -> out: M=256, N=1024, K=1024
  gemm_pw_kernel<<<dim3(C_DIM / 16, S_DIM / 256, TB), 256, 0, stream>>>(
      ff2_pw_w, (long)C_DIM * F_DIM, B_DIM, ff2_pw_b, (long)C_DIM,
      dwbuf, (long)F_DIM * S_DIM, xn, (long)C_DIM * S_DIM,
      out, (long)C_DIM * S_DIM, C_DIM, S_DIM, F_DIM, 0);
}